// Sobel_34763465294602
// MI455X (gfx1250) — compile-verified
//
#include <hip/hip_runtime.h>
#include <math.h>
#include <stdint.h>

// Sobel 3x3 + magnitude + border-zero on [32,1,1024,1024] fp32.
// Memory-bound (256 MiB round trip ~11us at 23.3 TB/s). Strategy:
//   - stage input tile (+halo) into LDS once via gfx1250 async global->LDS DMA
//     (GLOBAL_LOAD_ASYNC_TO_LDS_B128, tracked by ASYNCcnt),
//   - 3x3 reuse out of LDS with register sliding window down the column,
//   - coalesced fp32 stores.
// Expected gfx1250-specific codegen: global_load_async_to_lds_b128,
// s_wait_asynccnt, ds_load_*, s_barrier_signal/wait.

#define IMG_W 1024
#define IMG_H 1024
#define TILE_W 256
#define TILE_H 32
#define LDS_STRIDE 272            // floats per staged row (264 used + 8 pad; 272 % 64 = 16 banks)
#define LDS_ROWS   (TILE_H + 2)   // 34
#define ROW_SLOTS  66             // float4 slots per staged row (264 floats: cols x0-4 .. x0+259)
#define N_SLOTS    (LDS_ROWS * ROW_SLOTS)  // 2244

__global__ __launch_bounds__(256)
void sobel_mag_kernel(const float* __restrict__ img, float* __restrict__ out)
{
    __shared__ float tile[LDS_STRIDE * LDS_ROWS];

    const int t  = threadIdx.x;
    const int x0 = blockIdx.x * TILE_W;   // tile origin (output coords)
    const int y0 = blockIdx.y * TILE_H;
    const int b  = blockIdx.z;

    const float* base = img + (size_t)b * (IMG_H * IMG_W);
    const unsigned long long sbase = (unsigned long long)(uintptr_t)base;

    // ---------------- stage tile + halo into LDS (async DMA) ----------------
    // Staged region: rows y0-1 .. y0+TILE_H, cols x0-4 .. x0+259 (16B-aligned
    // slots). Addresses are clamped into the image so nothing faults; clamped
    // halo slots hold finite garbage that only feeds border pixels, which are
    // masked to zero below.
    #pragma unroll
    for (int i = 0; i < 9; ++i) {
        int idx = t + i * 256;
        if (idx < N_SLOTS) {
            int r  = idx / ROW_SLOTS;
            int c  = idx - r * ROW_SLOTS;
            int gy = y0 - 1 + r;
            gy = (gy < 0) ? 0 : ((gy > IMG_H - 1) ? IMG_H - 1 : gy);
            int gc = x0 - 4 + 4 * c;
            gc = (gc < 0) ? 0 : ((gc > IMG_W - 4) ? IMG_W - 4 : gc);
            unsigned voff = (unsigned)((gy * IMG_W + gc) * 4);      // byte offset (<4 MiB)
            unsigned lds  = (unsigned)(uintptr_t)(&tile[r * LDS_STRIDE + 4 * c]);
            asm volatile("global_load_async_to_lds_b128 %0, %1, %2 offset:0"
                         :: "v"(lds), "v"(voff), "s"(sbase) : "memory");
        }
    }
#if defined(__has_builtin) && __has_builtin(__builtin_amdgcn_s_wait_asynccnt)
    __builtin_amdgcn_s_wait_asynccnt(0);
#else
    asm volatile("s_wait_asynccnt 0" ::: "memory");
#endif
    __syncthreads();

    // ---------------- compute: one column per thread, 32 rows ----------------
    // LDS col for global col g is (g - (x0 - 4)); center for local col lx is lx+4.
    const int lx    = t;
    const int cbase = lx + 3;                 // leftmost tap (center-1)
    const int X     = x0 + lx;
    const float xmask = (X > 0 && X < IMG_W - 1) ? 1.0f : 0.0f;

    float h0, s0, h1, s1;
    {
        float a = tile[0 * LDS_STRIDE + cbase];
        float m = tile[0 * LDS_STRIDE + cbase + 1];
        float d = tile[0 * LDS_STRIDE + cbase + 2];
        h0 = a - d;
        s0 = a + 2.0f * m + d;
        a = tile[1 * LDS_STRIDE + cbase];
        m = tile[1 * LDS_STRIDE + cbase + 1];
        d = tile[1 * LDS_STRIDE + cbase + 2];
        h1 = a - d;
        s1 = a + 2.0f * m + d;
    }

    float* orow = out + (size_t)b * (IMG_H * IMG_W) + (size_t)y0 * IMG_W + X;

    #pragma unroll 4
    for (int r = 0; r < TILE_H; ++r) {
        const int j = r + 2;
        float a = tile[j * LDS_STRIDE + cbase];
        float m = tile[j * LDS_STRIDE + cbase + 1];
        float d = tile[j * LDS_STRIDE + cbase + 2];
        float h2 = a - d;
        float s2 = a + 2.0f * m + d;

        float gx = 2.0f * (h0 + h2) + 4.0f * h1;   // Gx cross-correlation
        float gy = 2.0f * (s0 - s2);               // Gy cross-correlation
        float mag = sqrtf(gx * gx + gy * gy);

        const int Y = y0 + r;
        const float msk = (Y > 0 && Y < IMG_H - 1) ? xmask : 0.0f;
        orow[(size_t)r * IMG_W] = mag * msk;

        h0 = h1; h1 = h2;
        s0 = s1; s1 = s2;
    }
}

extern "C" void kernel_launch(void* const* d_in, const int* in_sizes, int n_in,
                              void* d_out, int out_size, void* d_ws, size_t ws_size,
                              hipStream_t stream)
{
    const float* img = (const float*)d_in[0];   // [B,1,1024,1024] fp32
    float* out = (float*)d_out;                 // [B,1,1024,1024] fp32
    const int B = in_sizes[0] / (IMG_H * IMG_W);

    dim3 grid(IMG_W / TILE_W, IMG_H / TILE_H, B);
    dim3 block(256);
    sobel_mag_kernel<<<grid, block, 0, stream>>>(img, out);
}